// HeteroHyperConvNetwork_20358144983739
// MI455X (gfx1250) — compile-verified
//
#include <hip/hip_runtime.h>

#define NPOI   100000
#define NEDGE  50000
#define NNZC   3200000
#define DIM    128

typedef __attribute__((ext_vector_type(2))) float v2f;
typedef __attribute__((ext_vector_type(8))) float v8f;

#define WMMA_F32(A, B, C) \
  __builtin_amdgcn_wmma_f32_16x16x4_f32(false, (A), false, (B), (short)0, (C), false, false)

// ---------------------------------------------------------------------------
// WMMA GEMM: Y[M,N] = X[M,K] @ W[N,K]^T   (row-major, arbitrary row strides)
// One wave32 computes a 32x64 tile: 2 m-tiles x 4 n-tiles of f32 WMMA.
// K is a template constant -> full unroll + software-pipelined operand
// prefetch (next k-step's 6 loads issued before current 8 WMMAs).
// Tail handling: if the second m-tile is out of range, its row pointer is
// clamped to a valid row (loads always legal, no per-iteration branches);
// its results are simply not stored.
// A 16x4 layout: lanes 0-15 -> M=lane, K={0,1}; lanes 16-31 -> M=lane-16, K={2,3}
// B 4x16 layout: lanes 0-15 -> N=lane, K={0,1}; lanes 16-31 -> N=lane-16, K={2,3}
// C/D: vgpr v -> row M = v + 8*(lane>=16), col N = lane&15.
// Requires M%16==0, N%64==0 (holds at all call sites).
// ---------------------------------------------------------------------------
template <int K>
__global__ void gemm_xwt_wmma(const float* __restrict__ X, int ldx,
                              const float* __restrict__ W, int ldw,
                              float* __restrict__ Y, int ldy,
                              int M, int N) {
  const int wpb  = blockDim.x >> 5;
  const int wave = blockIdx.x * wpb + (threadIdx.x >> 5);
  const int ntChunks = N >> 6;                 // 64 output columns per wave
  const int mChunks  = (M + 31) >> 5;          // 32 output rows per wave
  if (wave >= mChunks * ntChunks) return;      // whole-wave uniform exit
  const int mc   = wave / ntChunks;
  const int nt   = wave - mc * ntChunks;
  const int lane = threadIdx.x & 31;
  const int half = lane >> 4;
  const int l16  = lane & 15;
  const bool has2 = (mc * 32 + 16) < M;        // second m-tile valid?
  // clamp tile-1 rows into range so its loads are always legal
  const int r1 = has2 ? (mc * 32 + 16 + l16) : (mc * 32 + l16);

  const float* xr0 = X + (size_t)(mc * 32 + l16) * ldx + 2 * half;
  const float* xr1 = X + (size_t)r1 * ldx + 2 * half;
  const float* wr  = W + (size_t)(nt * 64 + l16) * ldw + 2 * half;

  v8f c00 = {}, c01 = {}, c02 = {}, c03 = {};
  v8f c10 = {}, c11 = {}, c12 = {}, c13 = {};

  // prologue: load k-step 0
  v2f a0 = *(const v2f*)xr0;
  v2f a1 = *(const v2f*)xr1;
  v2f b0 = *(const v2f*)(wr);
  v2f b1 = *(const v2f*)(wr + 16 * ldw);
  v2f b2 = *(const v2f*)(wr + 32 * ldw);
  v2f b3 = *(const v2f*)(wr + 48 * ldw);

#pragma unroll
  for (int kb = 0; kb < K; kb += 4) {
    // software pipeline: prefetch next k-step before consuming current one
    v2f na0 = a0, na1 = a1, nb0 = b0, nb1 = b1, nb2 = b2, nb3 = b3;
    if (kb + 4 < K) {
      const int kn = kb + 4;
      na0 = *(const v2f*)(xr0 + kn);
      na1 = *(const v2f*)(xr1 + kn);
      nb0 = *(const v2f*)(wr + kn);
      nb1 = *(const v2f*)(wr + 16 * ldw + kn);
      nb2 = *(const v2f*)(wr + 32 * ldw + kn);
      nb3 = *(const v2f*)(wr + 48 * ldw + kn);
    }
    c00 = WMMA_F32(a0, b0, c00);
    c01 = WMMA_F32(a0, b1, c01);
    c02 = WMMA_F32(a0, b2, c02);
    c03 = WMMA_F32(a0, b3, c03);
    c10 = WMMA_F32(a1, b0, c10);
    c11 = WMMA_F32(a1, b1, c11);
    c12 = WMMA_F32(a1, b2, c12);
    c13 = WMMA_F32(a1, b3, c13);
    a0 = na0; a1 = na1; b0 = nb0; b1 = nb1; b2 = nb2; b3 = nb3;
  }

  float* yb0 = Y + (size_t)(mc * 32 + 8 * half) * ldy + nt * 64 + l16;
#pragma unroll
  for (int v = 0; v < 8; ++v) {
    yb0[(size_t)v * ldy +  0] = c00[v];
    yb0[(size_t)v * ldy + 16] = c01[v];
    yb0[(size_t)v * ldy + 32] = c02[v];
    yb0[(size_t)v * ldy + 48] = c03[v];
  }
  if (has2) {
    float* yb1 = yb0 + (size_t)16 * ldy;
#pragma unroll
    for (int v = 0; v < 8; ++v) {
      yb1[(size_t)v * ldy +  0] = c10[v];
      yb1[(size_t)v * ldy + 16] = c11[v];
      yb1[(size_t)v * ldy + 32] = c12[v];
      yb1[(size_t)v * ldy + 48] = c13[v];
    }
  }
}

// ---------------------------------------------------------------------------
// COO SpMM scatter: one wave per nonzero. Lane l handles cols [4l, 4l+4).
// Y[rows[i], 0:128] += vals[i] * X[cols[i], 0:128]   via f32 HW atomics.
// ---------------------------------------------------------------------------
__global__ void spmm_scatter(const int* __restrict__ rows, const int* __restrict__ cols,
                             const float* __restrict__ vals,
                             const float* __restrict__ X, int ldx,
                             float* __restrict__ Y, int ldy, int nnz) {
  const int wave = blockIdx.x * (blockDim.x >> 5) + (threadIdx.x >> 5);
  if (wave >= nnz) return;
  const int lane = threadIdx.x & 31;
  const int r = rows[wave];
  const int c = cols[wave];
  const float v = vals[wave];
  float4 d = ((const float4*)(X + (size_t)c * ldx))[lane];
  float* y = Y + (size_t)r * ldy + lane * 4;
  unsafeAtomicAdd(y + 0, v * d.x);
  unsafeAtomicAdd(y + 1, v * d.y);
  unsafeAtomicAdd(y + 2, v * d.z);
  unsafeAtomicAdd(y + 3, v * d.w);
}

// ------------------------- elementwise helpers -----------------------------
__global__ void copy2_f4(const float* __restrict__ s, float* __restrict__ d1,
                         float* __restrict__ d2, int n4) {
  int i = blockIdx.x * blockDim.x + threadIdx.x;
  if (i < n4) {
    float4 v = ((const float4*)s)[i];
    ((float4*)d1)[i] = v;
    ((float4*)d2)[i] = v;
  }
}
__global__ void copy1_f4(const float* __restrict__ s, float* __restrict__ d, int n4) {
  int i = blockIdx.x * blockDim.x + threadIdx.x;
  if (i < n4) ((float4*)d)[i] = ((const float4*)s)[i];
}
__global__ void zero_rows128(float* __restrict__ Y, int ldy, int nrows) {
  int row = blockIdx.x * (blockDim.x >> 5) + (threadIdx.x >> 5);
  if (row >= nrows) return;
  float4 z = make_float4(0.f, 0.f, 0.f, 0.f);
  ((float4*)(Y + (size_t)row * ldy))[threadIdx.x & 31] = z;
}
__global__ void add_into_f4(float* __restrict__ dst, const float* __restrict__ src, int n4) {
  int i = blockIdx.x * blockDim.x + threadIdx.x;
  if (i < n4) {
    float4 a = ((float4*)dst)[i];
    float4 b = ((const float4*)src)[i];
    a.x += b.x; a.y += b.y; a.z += b.z; a.w += b.w;
    ((float4*)dst)[i] = a;
  }
}
// ecur += fus; esum += ecur(new)
__global__ void update_e_f4(float* __restrict__ ecur, float* __restrict__ esum,
                            const float* __restrict__ fus, int n4) {
  int i = blockIdx.x * blockDim.x + threadIdx.x;
  if (i < n4) {
    float4 e = ((float4*)ecur)[i];
    float4 f = ((const float4*)fus)[i];
    e.x += f.x; e.y += f.y; e.z += f.z; e.w += f.w;
    ((float4*)ecur)[i] = e;
    float4 s = ((float4*)esum)[i];
    s.x += e.x; s.y += e.y; s.z += e.z; s.w += e.w;
    ((float4*)esum)[i] = s;
  }
}
__global__ void scale_f4(const float* __restrict__ src, float* __restrict__ dst,
                         float s, int n4) {
  int i = blockIdx.x * blockDim.x + threadIdx.x;
  if (i < n4) {
    float4 v = ((const float4*)src)[i];
    v.x *= s; v.y *= s; v.z *= s; v.w *= s;
    ((float4*)dst)[i] = v;
  }
}

// ---------------------------------------------------------------------------
extern "C" void kernel_launch(void* const* d_in, const int* in_sizes, int n_in,
                              void* d_out, int out_size, void* d_ws, size_t ws_size,
                              hipStream_t stream) {
  const float* poi     = (const float*)d_in[0];
  const float* edg     = (const float*)d_in[1];
  const float* Wpoi    = (const float*)d_in[2];
  const float* Wedge   = (const float*)d_in[3];
  const float* Wfus    = (const float*)d_in[4];
  const int*   p2e_row = (const int*)d_in[5];
  const int*   p2e_col = (const int*)d_in[6];
  const float* p2e_val = (const float*)d_in[7];
  const int*   e2p_row = (const int*)d_in[8];
  const int*   e2p_col = (const int*)d_in[9];
  const float* e2p_val = (const float*)d_in[10];
  float* out = (float*)d_out;

  const size_t PO = (size_t)NPOI * DIM;     // 12.8M floats
  const size_t ED = (size_t)NEDGE * DIM;    //  6.4M floats

  float* w = (float*)d_ws;
  float* p_a   = w; w += PO;                 // current p
  float* p_b   = w; w += PO;                 // scratch: pW, then p_next
  float* p_sum = w; w += PO;
  float* e_cur = w; w += ED;
  float* e_sum = w; w += ED;
  float* fin   = w; w += 2 * ED;             // [N_EDGE, 256] concat buffer
  float* fus   = w; w += ED;                 // fused output [N_EDGE, 128]

  const int PO4 = (int)(PO / 4);             // 3,200,000
  const int ED4 = (int)(ED / 4);             // 1,600,000

  // init accumulators: p_sum/e_sum start at layer-0 embeddings
  copy2_f4<<<PO4 / 256, 256, 0, stream>>>(poi, p_a, p_sum, PO4);
  copy2_f4<<<ED4 / 256, 256, 0, stream>>>(edg, e_cur, e_sum, ED4);

  auto gemm_blocks = [](int M, int N) {
    int tiles = ((M + 31) >> 5) * (N >> 6);
    return (tiles + 7) / 8;                  // 8 waves per 256-thread block
  };

  for (int layer = 0; layer < 2; ++layer) {
    // 1) pW = p @ W_poi^T  -> p_b  [N_POI,128]
    gemm_xwt_wmma<128><<<gemm_blocks(NPOI, DIM), 256, 0, stream>>>(
        p_a, DIM, Wpoi, DIM, p_b, DIM, NPOI, DIM);
    // 2) clear scatter target: fin[:, 0:128]
    zero_rows128<<<NEDGE / 8, 256, 0, stream>>>(fin, 2 * DIM, NEDGE);
    // 3) poi_msg: fin[:,0:128] += p2e * pW
    spmm_scatter<<<NNZC / 8, 256, 0, stream>>>(p2e_row, p2e_col, p2e_val,
                                               p_b, DIM, fin, 2 * DIM, NNZC);
    // 4) edge_msg: fin[:,128:256] = e @ W_edge^T
    gemm_xwt_wmma<128><<<gemm_blocks(NEDGE, DIM), 256, 0, stream>>>(
        e_cur, DIM, Wedge, DIM, fin + DIM, 2 * DIM, NEDGE, DIM);
    // 5) fused = [poi_msg | edge_msg] @ W_fus^T   (K = 256)
    gemm_xwt_wmma<256><<<gemm_blocks(NEDGE, DIM), 256, 0, stream>>>(
        fin, 2 * DIM, Wfus, 2 * DIM, fus, DIM, NEDGE, DIM);
    // 6) p_next = p_prev + e2p*fused : seed with p_prev, scatter-add
    copy1_f4<<<PO4 / 256, 256, 0, stream>>>(p_a, p_b, PO4);
    spmm_scatter<<<NNZC / 8, 256, 0, stream>>>(e2p_row, e2p_col, e2p_val,
                                               fus, DIM, p_b, DIM, NNZC);
    // 7) e = fused + e_prev ; running sums
    update_e_f4<<<ED4 / 256, 256, 0, stream>>>(e_cur, e_sum, fus, ED4);
    add_into_f4<<<PO4 / 256, 256, 0, stream>>>(p_sum, p_b, PO4);
    // swap p buffers
    float* t = p_a; p_a = p_b; p_b = t;
  }

  // mean over {init, layer1, layer2} = sum / 3
  const float inv3 = 1.0f / 3.0f;
  scale_f4<<<PO4 / 256, 256, 0, stream>>>(p_sum, out, inv3, PO4);
  scale_f4<<<ED4 / 256, 256, 0, stream>>>(e_sum, out + PO, inv3, ED4);
}